// PCHIPKANLayer_5282809774968
// MI455X (gfx1250) — compile-verified
//
#include <hip/hip_runtime.h>

typedef __attribute__((ext_vector_type(2))) float v2f;
typedef __attribute__((ext_vector_type(8))) float v8f;

#define D_IN    256
#define D_OUT   256
#define KNOTS   64
#define NB      1024
#define NCHUNK  16
#define I_PER_CHUNK (D_IN / NCHUNK)
#define STEP    (4.0f / 63.0f)
#define EPSF    1e-12f
#define YS_ELEMS ((size_t)D_IN * KNOTS * D_OUT * 2)   // interleaved {y,slope}: 8,388,608 floats (32 MB)

// ---------------------------------------------------------------------------
// Kernel 1: PCHIP slopes + transpose
//   coeffs[o][i][k] -> YS[i][k][o][2] = {y, slope}   (o-contiguous float2 rows)
// ---------------------------------------------------------------------------
__global__ void pchip_prep_kernel(const float* __restrict__ coeffs,
                                  float* __restrict__ YS) {
    const int i  = blockIdx.x;          // input dim
    const int o0 = blockIdx.y * 16;     // output tile base
    __shared__ float yt[16][KNOTS + 1]; // +1 pad: conflict-free strided reads

    const int t = threadIdx.x;          // 256 threads
    #pragma unroll
    for (int p = 0; p < 4; ++p) {
        int e  = p * 256 + t;
        int oo = e >> 6;
        int k  = e & 63;
        yt[oo][k] = coeffs[(size_t)(o0 + oo) * (D_IN * KNOTS) + (size_t)i * KNOTS + k];
    }
    __syncthreads();

    const int oo = t & 15;
    const int k0 = t >> 4;
    #pragma unroll
    for (int j = 0; j < 4; ++j) {
        int k = k0 + 16 * j;
        float yk = yt[oo][k];
        float dprev = (k >= 1)  ? (yk - yt[oo][k - 1]) / (STEP + EPSF) : 0.0f;
        float dnext = (k <= 62) ? (yt[oo][k + 1] - yk) / (STEP + EPSF) : 0.0f;
        float d;
        if (k == 0) {
            d = dnext;                      // d_first = delta[0]
        } else if (k == 63) {
            d = dprev;                      // d_last = delta[K-2]
        } else {
            bool ss = (dprev * dnext) > 0.0f;
            const float w1 = 3.0f * STEP;   // 2*h + h (uniform knots)
            const float w2 = 3.0f * STEP;   // h + 2*h
            d = ss ? (w1 + w2) / (w1 / (dprev + EPSF) + w2 / (dnext + EPSF) + EPSF)
                   : 0.0f;
        }
        size_t off = (((size_t)i * KNOTS + k) * D_OUT + (size_t)(o0 + oo)) * 2;
        v2f ys; ys.x = yk; ys.y = d;
        *(v2f*)(YS + off) = ys;             // 128B contiguous per 16 lanes
    }
}

// ---------------------------------------------------------------------------
// Kernel 2: rank-4 WMMA accumulation, static 16-group unroll.
//   block = 256 threads (8 waves); grid = (NB/16 sample-tiles, NCHUNK i-chunks)
//   wave w owns output cols [32w, 32w+32): two 16x16 f32 accumulators.
// ---------------------------------------------------------------------------
__global__ void pchip_wmma_kernel(const float* __restrict__ x,
                                  const float* __restrict__ YS,
                                  float* __restrict__ part) {
    const int m     = blockIdx.x;             // 16-sample tile
    const int chunk = blockIdx.y;             // i-chunk
    const int wave  = threadIdx.x >> 5;       // 0..7
    const int lane  = threadIdx.x & 31;
    const int n0    = wave * 32;              // output-column base (2 tiles)
    const int s     = lane & 15;              // my sample within tile
    const int half  = lane >> 4;              // 0: rows K=0,1   1: rows K=2,3
    const int b     = m * 16 + s;             // global sample

    // loop-invariant per-lane element offset into YS
    const int lanepart = (half * D_OUT + n0 + s) * 2;

    v8f acc0 = {};
    v8f acc1 = {};

    const int ibeg = chunk * I_PER_CHUNK;
    for (int i = ibeg; i < ibeg + I_PER_CHUNK; ++i) {
        // per-sample Hermite weights (each lane computes its own sample's,
        // so the A fragment needs no cross-lane shuffles)
        float xv = x[(size_t)b * D_IN + i];
        float xc = fminf(2.0f, fmaxf(-2.0f, xv));
        int   idx = (int)floorf((xc + 2.0f) * (63.0f / 4.0f));
        idx = idx < 0 ? 0 : (idx > 62 ? 62 : idx);
        float x0 = -2.0f + (float)idx * STEP;
        float h  = STEP + EPSF;
        float tt = (xc - x0) / h;
        float t2 = tt * tt, t3 = t2 * tt;
        float w0 = 2.0f * t3 - 3.0f * t2 + 1.0f;   // * y0
        float w1 = (t3 - 2.0f * t2 + tt) * h;      // * d0
        float w2 = -2.0f * t3 + 3.0f * t2;         // * y1
        float w3 = (t3 - t2) * h;                  // * d1
        bool below = xv < -2.0f, above = xv > 2.0f;
        if (below) { w0 = 1.0f; w1 = xv + 2.0f; w2 = 0.0f; w3 = 0.0f; } // idx==0
        if (above) { w0 = 0.0f; w1 = 0.0f; w2 = 1.0f; w3 = xv - 2.0f; } // idx==62
        const float myA0 = half ? w2 : w0;  // A v0: K=0 (lanes<16) / K=2 (lanes>=16)
        const float myA1 = half ? w3 : w1;  // A v1: K=1            / K=3

        const float* Ybase = YS + (size_t)i * (KNOTS * D_OUT * 2);

        // one group per sample; fully unrolled, uniform control flow,
        // EXEC stays all-ones (WMMA requirement)
        #pragma unroll
        for (int g = 0; g < 16; ++g) {
            int cur = __builtin_amdgcn_readlane(idx, g);   // uniform interval
            bool ing = (s == g);
            v2f a;
            a.x = ing ? myA0 : 0.0f;
            a.y = ing ? myA1 : 0.0f;

            // B fragments: lanes<16 -> row cur ({y,s}); lanes>=16 -> row cur+1
            const float* p = Ybase + (size_t)cur * (D_OUT * 2) + lanepart;
            v2f b0 = *(const v2f*)(p);        // cols n0   .. n0+15
            v2f b1 = *(const v2f*)(p + 32);   // cols n0+16.. n0+31 (+128B)

            acc0 = __builtin_amdgcn_wmma_f32_16x16x4_f32(
                false, a, false, b0, (short)0, acc0, false, false);
            acc1 = __builtin_amdgcn_wmma_f32_16x16x4_f32(
                false, a, false, b1, (short)0, acc1, false, false);
        }
    }

    // C/D layout: VGPR r -> M = r + 8*half, N = lane&15
    float* pbase = part + ((size_t)chunk * NB + (size_t)m * 16) * D_OUT;
    #pragma unroll
    for (int r = 0; r < 8; ++r) {
        int row = r + 8 * half;
        pbase[(size_t)row * D_OUT + (size_t)(n0 + s)]      = acc0[r];
        pbase[(size_t)row * D_OUT + (size_t)(n0 + 16 + s)] = acc1[r];
    }
}

// ---------------------------------------------------------------------------
// Kernel 3: reduce chunk partials + bias -> out
// ---------------------------------------------------------------------------
__global__ void pchip_reduce_kernel(const float* __restrict__ part,
                                    const float* __restrict__ bias,
                                    float* __restrict__ out) {
    int e = blockIdx.x * 256 + threadIdx.x;   // < NB*D_OUT
    float sum = bias[e & (D_OUT - 1)];
    #pragma unroll
    for (int c = 0; c < NCHUNK; ++c)
        sum += part[(size_t)c * NB * D_OUT + e];
    out[e] = sum;
}

// ---------------------------------------------------------------------------
extern "C" void kernel_launch(void* const* d_in, const int* in_sizes, int n_in,
                              void* d_out, int out_size, void* d_ws, size_t ws_size,
                              hipStream_t stream) {
    (void)in_sizes; (void)n_in; (void)out_size; (void)ws_size;
    const float* x      = (const float*)d_in[0];  // (1024, 256)
    const float* coeffs = (const float*)d_in[1];  // (256, 256, 64)
    const float* bias   = (const float*)d_in[2];  // (256,)
    // d_in[3] = knots: uniform linspace(-2,2,64) by construction; constants used.

    float* YS   = (float*)d_ws;            // [256][64][256][2]  (32 MB)
    float* part = YS + YS_ELEMS;           // [NCHUNK][1024][256] (16 MB)
    float* out  = (float*)d_out;

    pchip_prep_kernel<<<dim3(D_IN, D_OUT / 16), 256, 0, stream>>>(coeffs, YS);
    pchip_wmma_kernel<<<dim3(NB / 16, NCHUNK), 256, 0, stream>>>(x, YS, part);
    pchip_reduce_kernel<<<dim3((NB * D_OUT) / 256), 256, 0, stream>>>(part, bias, out);
}